// CenterLoss_2585570312419
// MI455X (gfx1250) — compile-verified
//
#include <hip/hip_runtime.h>
#include <hip/hip_bf16.h>

typedef __attribute__((ext_vector_type(2))) float v2f;
typedef __attribute__((ext_vector_type(8))) float v8f;

#define NB      8192     // batch
#define DFEAT   512      // feature dim
#define NTILES  (NB / 16)

// One wave32 per 16-row tile. K advances by 8 per iteration: the low half-wave
// (lanes 0-15, rows 0-15) loads float4 at k..k+3, the high half-wave loads
// float4 at k+4..k+7 for the same rows. Since the row-sum is k-permutation
// invariant and B is all-ones, any assignment of k-values to A slots is valid:
// WMMA#1 takes elements {0,1} of each half's float4, WMMA#2 takes {2,3}.
__global__ __launch_bounds__(256) void center_loss_wmma(
    const float* __restrict__ x,
    const int*   __restrict__ labels,
    const float* __restrict__ centers,
    float*       __restrict__ partial)
{
    const int wave = threadIdx.x >> 5;
    const int lane = threadIdx.x & 31;
    const int tile = blockIdx.x * 8 + wave;          // 0..511
    const int rlo  = lane & 15;
    const int row  = tile * 16 + rlo;
    const int kOff = (lane >> 4) << 2;               // 0 for lanes 0-15, 4 for 16-31

    const int lbl = labels[row];
    const float* xr = x       + (size_t)row * DFEAT + kOff;
    const float* cr = centers + (size_t)lbl * DFEAT + kOff;

    v2f ones; ones[0] = 1.0f; ones[1] = 1.0f;        // B matrix: all 1.0 -> layout-proof
    v8f acc = {};                                    // C/D accumulator (zeros)

    #pragma unroll 4
    for (int i = 0; i < DFEAT / 8; ++i) {
        const float4 xv = *reinterpret_cast<const float4*>(xr + 8 * i);
        const float4 cv = *reinterpret_cast<const float4*>(cr + 8 * i);
        float d0 = xv.x - cv.x;
        float d1 = xv.y - cv.y;
        float d2 = xv.z - cv.z;
        float d3 = xv.w - cv.w;
        v2f a1; a1[0] = d0 * d0; a1[1] = d1 * d1;
        v2f a2; a2[0] = d2 * d2; a2[1] = d3 * d3;
        // 8 args: (neg_a, A, neg_b, B, c_mod, C, reuse_a, reuse_b)
        acc = __builtin_amdgcn_wmma_f32_16x16x4_f32(
                  false, a1, false, ones, (short)0, acc, false, false);
        acc = __builtin_amdgcn_wmma_f32_16x16x4_f32(
                  false, a2, false, ones, (short)0, acc, false, false);
    }

    // C/D layout (f32 16x16): VGPR v -> row v (lanes 0-15) / row v+8 (lanes 16-31).
    // Clamp each per-sample distance, then sum the 8 rows this half-wave owns.
    float s = 0.0f;
    #pragma unroll
    for (int v = 0; v < 8; ++v) {
        float d = acc[v];
        d = fminf(fmaxf(d, 1e-12f), 1e12f);
        s += d;
    }
    // Combine rows 0..7 (lower half) with rows 8..15 (upper half).
    float other = __shfl_xor(s, 16, 32);
    if (lane == 0) partial[tile] = s + other;
}

// Deterministic tree reduction of the 512 tile partials; scale by 1/B.
__global__ __launch_bounds__(256) void reduce_partials(
    const float* __restrict__ partial, float* __restrict__ out)
{
    __shared__ float sm[256];
    float s = 0.0f;
    for (int i = threadIdx.x; i < NTILES; i += 256) s += partial[i];
    sm[threadIdx.x] = s;
    __syncthreads();
    #pragma unroll
    for (int w = 128; w > 0; w >>= 1) {
        if (threadIdx.x < w) sm[threadIdx.x] += sm[threadIdx.x + w];
        __syncthreads();
    }
    if (threadIdx.x == 0) out[0] = sm[0] * (1.0f / (float)NB);
}

extern "C" void kernel_launch(void* const* d_in, const int* in_sizes, int n_in,
                              void* d_out, int out_size, void* d_ws, size_t ws_size,
                              hipStream_t stream) {
    (void)in_sizes; (void)n_in; (void)out_size; (void)ws_size;
    const float* x       = (const float*)d_in[0];
    const int*   labels  = (const int*)d_in[1];
    const float* centers = (const float*)d_in[2];
    float* out     = (float*)d_out;
    float* partial = (float*)d_ws;   // 512 floats of scratch

    center_loss_wmma<<<NTILES / 8, 256, 0, stream>>>(x, labels, centers, partial);
    reduce_partials<<<1, 256, 0, stream>>>(partial, out);
}